// StockForecastTemporalConvGNN_3332894621906
// MI455X (gfx1250) — compile-verified
//
#include <hip/hip_runtime.h>

typedef __attribute__((ext_vector_type(2))) float v2f;
typedef __attribute__((ext_vector_type(8))) float v8f;
typedef __attribute__((ext_vector_type(4))) unsigned int u32x4;
typedef __attribute__((ext_vector_type(8))) int i32x8;
typedef __attribute__((ext_vector_type(4))) int i32x4;

#define B_N   2000
#define T_N   32
#define CIN_N 32
#define COUT_N 128
#define E_N   32000
#define NEG_S 0.01f
#define EPS_S 1e-5f
#define NROW  (T_N * B_N)         /* 64000 node-instances */
#define NF    (NROW * COUT_N)     /* 8,192,000 floats per feature buffer */

__device__ __forceinline__ float leaky(float v) { return v >= 0.f ? v : NEG_S * v; }

// ---------------------------------------------------------------- zero fill
__global__ void k_zero(float* __restrict__ p, int n) {
  int i = blockIdx.x * blockDim.x + threadIdx.x;
  if (i < n) p[i] = 0.f;
}

// ------------------------------------------- conv1d + LeakyReLU + LayerNorm
// One block per b (128 threads = one thread per output channel co).
// Writes directly in time-parallel layout xp[(t*B+b)*COUT + co].
__global__ void __launch_bounds__(128) k_conv_ln(
    const float* __restrict__ x, const float* __restrict__ cw,
    const float* __restrict__ cb, const float* __restrict__ lg,
    const float* __restrict__ lb, float* __restrict__ xp) {
  __shared__ float sx[CIN_N * T_N];
  __shared__ float rs[128];
  __shared__ float rq[128];
  __shared__ float s_mu, s_rstd;
  const int tid = threadIdx.x;
  const int b = blockIdx.x;
  for (int i = tid; i < CIN_N * T_N; i += 128) sx[i] = x[b * CIN_N * T_N + i];
  __syncthreads();

  const int co = tid;
  float y[T_N];
  const float bias = cb[co];
#pragma unroll
  for (int t = 0; t < T_N; ++t) y[t] = bias;
  for (int ci = 0; ci < CIN_N; ++ci) {
    const float w0 = cw[(co * CIN_N + ci) * 3 + 0];
    const float w1 = cw[(co * CIN_N + ci) * 3 + 1];
    const float w2 = cw[(co * CIN_N + ci) * 3 + 2];
    const float* xr = &sx[ci * T_N];
#pragma unroll
    for (int t = 0; t < T_N; ++t) {
      float acc = w2 * xr[t];                 // causal taps: t-2, t-1, t
      if (t >= 1) acc += w1 * xr[t - 1];
      if (t >= 2) acc += w0 * xr[t - 2];
      y[t] += acc;
    }
  }
  float s = 0.f, q = 0.f;
#pragma unroll
  for (int t = 0; t < T_N; ++t) {
    y[t] = leaky(y[t]);
    s += y[t];
    q += y[t] * y[t];
  }
  rs[tid] = s; rq[tid] = q;
  __syncthreads();
  for (int off = 64; off > 0; off >>= 1) {
    if (tid < off) { rs[tid] += rs[tid + off]; rq[tid] += rq[tid + off]; }
    __syncthreads();
  }
  if (tid == 0) {
    const float inv = 1.f / (float)(COUT_N * T_N);
    float mu = rs[0] * inv;
    float var = rq[0] * inv - mu * mu;
    s_mu = mu;
    s_rstd = rsqrtf(var + EPS_S);
  }
  __syncthreads();
  const float mu = s_mu, rstd = s_rstd;
  for (int t = 0; t < T_N; ++t) {
    float v = (y[t] - mu) * rstd * lg[co * T_N + t] + lb[co * T_N + t];
    xp[(size_t)(t * B_N + b) * COUT_N + co] = v;
  }
}

// ------------------------------------------------ degree + edge gcn norm
// deg/norm are identical for every timestep replica -> compute once over E.
__global__ void k_degree(const int* __restrict__ ei, const float* __restrict__ ew,
                         float* __restrict__ deg) {
  int e = blockIdx.x * blockDim.x + threadIdx.x;
  if (e < E_N) unsafeAtomicAdd(&deg[ei[E_N + e]], ew[e]);
}

__global__ void k_edge_norm(const int* __restrict__ ei, const float* __restrict__ ew,
                            const float* __restrict__ deg, float* __restrict__ nrm) {
  int e = blockIdx.x * blockDim.x + threadIdx.x;
  if (e >= E_N) return;
  float ds = deg[ei[e]];
  float dd = deg[ei[E_N + e]];
  float a = ds > 0.f ? rsqrtf(fmaxf(ds, 1e-12f)) : 0.f;
  float c = dd > 0.f ? rsqrtf(fmaxf(dd, 1e-12f)) : 0.f;
  nrm[e] = a * ew[e] * c;
}

// --------------------------------------------- one propagation hop (scatter)
// One wave per (t, e) edge instance; lane handles 4 channels (float4 gather,
// 4x native global f32 atomic adds into the destination row). Working set
// (~33 MB per feature buffer) lives in the 192 MB L2, so the atomics are
// L2-resident RMWs, not HBM round-trips.
__global__ void __launch_bounds__(256) k_hop(
    const float* __restrict__ hin, float* __restrict__ hout,
    const float* __restrict__ nrm, const int* __restrict__ ei) {
  const int gid = blockIdx.x * 8 + (threadIdx.x >> 5);   // edge-instance id
  const int lane = threadIdx.x & 31;
  const int t = gid / E_N;
  const int e = gid - t * E_N;
  const float w = nrm[e];
  const int srow = t * B_N + ei[e];
  const int drow = t * B_N + ei[E_N + e];
  const float4 v = ((const float4*)(hin + (size_t)srow * COUT_N))[lane];
  float* dstp = hout + (size_t)drow * COUT_N + lane * 4;
  unsafeAtomicAdd(dstp + 0, w * v.x);
  unsafeAtomicAdd(dstp + 1, w * v.y);
  unsafeAtomicAdd(dstp + 2, w * v.z);
  unsafeAtomicAdd(dstp + 3, w * v.w);
}

// ----------------------------------------------- WMMA f32 GEMM: out (+)= A*W
// A: [64000 x 128] row-major, W: [128 x 128] row-major (K rows x N cols).
// Block = 256 threads = 8 waves; whole W (64 KB) staged in LDS via the
// Tensor Data Mover (descriptor-driven DMA, TENSORcnt), falling back to a
// float4 copy if the TDM builtin is unavailable. Wave handles one 16-row
// tile across all 8 column tiles; 32 K-steps of V_WMMA_F32_16X16X4_F32
// (f32 precision preserved end-to-end).
__global__ void __launch_bounds__(256) k_gemm(
    const float* __restrict__ A, const float* __restrict__ W,
    float* __restrict__ out, int accumulate) {
  __shared__ float sW[COUT_N * COUT_N];
  const int tid = threadIdx.x;

#if __has_builtin(__builtin_amdgcn_tensor_load_to_lds) && \
    __has_builtin(__builtin_amdgcn_s_wait_tensorcnt)
  if ((tid >> 5) == 0) {            // one TDM issue per block (wave 0)
    // Generic LDS pointers carry the byte offset in the low 32 bits
    // (LDS aperture maps by discarding the upper bits, ISA 10.2).
    unsigned lds_off = (unsigned)(size_t)(void*)&sW[0];
    unsigned long long ga = (unsigned long long)(size_t)(const void*)W;
    // D# group0: count=1 | lds_addr | global_addr[56:0] | type=2
    u32x4 g0;
    g0.x = 1u;                                   // count=1, no gather
    g0.y = lds_off;                              // lds_addr
    g0.z = (unsigned)(ga & 0xFFFFFFFFull);       // global_addr[31:0]
    g0.w = (unsigned)((ga >> 32) & 0x01FFFFFFull) | (2u << 30); // [56:32]|type=2
    // D# group1: data_size=4B, 2-D tensor 128x128, tile 128x128, stride0=128
    i32x8 g1;
    g1[0] = 0x00020000;                          // wg_mask=0, data_size=2 (4B)
    g1[1] = (int)((128u & 0xFFFFu) << 16);       // tensor_dim0[15:0] @ bits63:48
    g1[2] = (int)((128u >> 16) | ((128u & 0xFFFFu) << 16)); // dim0 hi | dim1 lo
    g1[3] = (int)((128u >> 16) | (128u << 16));  // dim1 hi | tile_dim0=128
    g1[4] = (int)(128u);                         // tile_dim1=128, tile_dim2=0
    g1[5] = (int)(128u);                         // tensor_dim0_stride[31:0]
    g1[6] = 0;                                   // stride0 hi | stride1 lo
    g1[7] = 0;                                   // stride1 hi
    i32x4 g2 = {0, 0, 0, 0};                     // 2-D tensor: groups 2/3 unused
    i32x4 g3 = {0, 0, 0, 0};
    i32x8 g4 = {0, 0, 0, 0, 0, 0, 0, 0};         // 6-arg form (clang-23 lane)
    __builtin_amdgcn_tensor_load_to_lds(g0, g1, g2, g3, g4, 0);
    __builtin_amdgcn_s_wait_tensorcnt(0);
  }
#else
  {
    const float4* W4 = (const float4*)W;
    float4* sW4 = (float4*)sW;
    for (int i = tid; i < (COUT_N * COUT_N) / 4; i += 256) sW4[i] = W4[i];
  }
#endif
  __syncthreads();

  const int wave = tid >> 5;
  const int lane = tid & 31;
  const int half = lane >> 4;      // 0: lanes 0-15, 1: lanes 16-31
  const int l16 = lane & 15;
  const int row0 = (blockIdx.x * 8 + wave) * 16;

  v8f acc[8];
  if (accumulate) {
    // C/D layout: VGPR v <-> M = v + half*8, N = l16 (per 16x16 f32 map)
#pragma unroll
    for (int n = 0; n < 8; ++n) {
      const int col = n * 16 + l16;
#pragma unroll
      for (int v = 0; v < 8; ++v)
        acc[n][v] = out[(size_t)(row0 + v + half * 8) * COUT_N + col];
    }
  } else {
#pragma unroll
    for (int n = 0; n < 8; ++n) {
      v8f z = {0.f, 0.f, 0.f, 0.f, 0.f, 0.f, 0.f, 0.f};
      acc[n] = z;
    }
  }

  // A 16x4 f32 fragment: lanes 0-15 hold K={0,1}, lanes 16-31 hold K={2,3}
  const float* Arow = A + (size_t)(row0 + l16) * COUT_N;
  for (int kk = 0; kk < 32; ++kk) {
    const int kbase = kk * 4 + half * 2;
    v2f a;
    a.x = Arow[kbase + 0];
    a.y = Arow[kbase + 1];
#pragma unroll
    for (int n = 0; n < 8; ++n) {
      v2f bf;                       // B 4x16: rows striped like C/D
      bf.x = sW[(kbase + 0) * COUT_N + n * 16 + l16];
      bf.y = sW[(kbase + 1) * COUT_N + n * 16 + l16];
      acc[n] = __builtin_amdgcn_wmma_f32_16x16x4_f32(
          false, a, false, bf, (short)0, acc[n], false, false);
    }
  }

#pragma unroll
  for (int n = 0; n < 8; ++n) {
    const int col = n * 16 + l16;
#pragma unroll
    for (int v = 0; v < 8; ++v)
      out[(size_t)(row0 + v + half * 8) * COUT_N + col] = acc[n][v];
  }
}

// --------------------------------- residual + LeakyReLU + transpose to NCHW
__global__ void k_final(const float* __restrict__ xp, const float* __restrict__ out,
                        const float* __restrict__ tb, float* __restrict__ y) {
  size_t i = (size_t)blockIdx.x * blockDim.x + threadIdx.x;
  if (i >= (size_t)NF) return;
  const int row = (int)(i >> 7);
  const int co = (int)(i & 127);
  const int t = row / B_N;
  const int b = row - t * B_N;
  float o = out[i] + tb[co];
  float r = xp[i] + leaky(o);
  y[((size_t)b * COUT_N + co) * T_N + t] = r;
}

extern "C" void kernel_launch(void* const* d_in, const int* in_sizes, int n_in,
                              void* d_out, int out_size, void* d_ws, size_t ws_size,
                              hipStream_t stream) {
  const float* x  = (const float*)d_in[0];
  const int*   ei = (const int*)d_in[1];
  const float* ew = (const float*)d_in[2];
  const float* cw = (const float*)d_in[3];
  const float* cb = (const float*)d_in[4];
  const float* lg = (const float*)d_in[5];
  const float* lb = (const float*)d_in[6];
  const float* tw = (const float*)d_in[7];
  const float* tb = (const float*)d_in[8];
  float* y = (float*)d_out;
  (void)in_sizes; (void)n_in; (void)out_size; (void)ws_size;

  float* xp  = (float*)d_ws;      // [64000 x 128]
  float* h1  = xp + NF;           // hop-1 features
  float* h2  = h1 + NF;           // hop-2 features
  float* out = h2 + NF;           // GEMM accumulator
  float* deg = out + NF;          // [2000]
  float* nrm = deg + B_N;         // [32000]

  k_zero<<<(B_N + 255) / 256, 256, 0, stream>>>(deg, B_N);
  k_conv_ln<<<B_N, 128, 0, stream>>>(x, cw, cb, lg, lb, xp);
  k_degree<<<(E_N + 255) / 256, 256, 0, stream>>>(ei, ew, deg);
  k_edge_norm<<<(E_N + 255) / 256, 256, 0, stream>>>(ei, ew, deg, nrm);

  k_gemm<<<NROW / 128, 256, 0, stream>>>(xp, tw + 0 * COUT_N * COUT_N, out, 0);

  k_zero<<<(NF + 255) / 256, 256, 0, stream>>>(h1, NF);
  k_hop<<<(T_N * E_N) / 8, 256, 0, stream>>>(xp, h1, nrm, ei);
  k_gemm<<<NROW / 128, 256, 0, stream>>>(h1, tw + 1 * COUT_N * COUT_N, out, 1);

  k_zero<<<(NF + 255) / 256, 256, 0, stream>>>(h2, NF);
  k_hop<<<(T_N * E_N) / 8, 256, 0, stream>>>(h1, h2, nrm, ei);
  k_gemm<<<NROW / 128, 256, 0, stream>>>(h2, tw + 2 * COUT_N * COUT_N, out, 1);

  k_final<<<(NF + 255) / 256, 256, 0, stream>>>(xp, out, tb, y);
}